// TopKPool_67602785239067
// MI455X (gfx1250) — compile-verified
//
#include <hip/hip_runtime.h>

typedef float v2f __attribute__((ext_vector_type(2)));
typedef float v8f __attribute__((ext_vector_type(8)));

#define BB 128
#define KD 512     // K slices per batch
#define DD 768
#define NEG_INF_F (-1e30f)

#define ROWS_PER_BLOCK 128   // k-rows per block in scores_kernel
#define DC 32                // D-chunk staged in LDS
#define LDS_PITCH 36         // padded row pitch (dwords) -> conflict-free fragment reads

// ---------------------------------------------------------------------------
// Kernel 1: scores[b,k] = dot(E[b,k,:], w) + bias, masked.
// WMMA f32 16x16x4 GEMV: w broadcast into all 16 B-columns, so every column of
// the 16x16 accumulator equals the 16 row-scores. LDS-staged, coalesced loads.
// ---------------------------------------------------------------------------
__global__ __launch_bounds__(256) void scores_kernel(
    const float* __restrict__ emb, const int* __restrict__ mask,
    const float* __restrict__ w, const float* __restrict__ bias,
    float* __restrict__ scores)
{
    __shared__ float tile[ROWS_PER_BLOCK * LDS_PITCH];
    __shared__ float wl[DD];

    const int tid  = threadIdx.x;
    const int lane = tid & 31;
    const int wave = tid >> 5;              // 0..7, 16 k-rows each
    const int k0   = blockIdx.x * ROWS_PER_BLOCK;
    const int bi   = blockIdx.y;
    const float* erow = emb + ((size_t)bi * KD + k0) * DD;

    // stage scorer weight once (reused by all 128 rows of this block)
    for (int i = tid; i < DD; i += 256) wl[i] = w[i];

    const int hi      = lane >> 4;          // 0: K=0,1   1: K=2,3
    const int m       = lane & 15;          // row within the wave's 16-row tile
    const int rowBase = wave * 16;

    v8f acc = {};

    for (int d0 = 0; d0 < DD; d0 += DC) {
        __syncthreads();
        // cooperative fill: 128 rows x 32 cols = 1024 float4, 4 per thread,
        // fully coalesced 128B requests per half-wave group of 8 lanes
        for (int j = 0; j < 4; ++j) {
            int idx = j * 256 + tid;        // 0..1023
            int r   = idx >> 3;
            int c4  = idx & 7;
            const float4 v = *(const float4*)(erow + (size_t)r * DD + d0 + c4 * 4);
            float* dst = &tile[r * LDS_PITCH + c4 * 4];
            dst[0] = v.x; dst[1] = v.y; dst[2] = v.z; dst[3] = v.w;
        }
        __syncthreads();

        for (int dd = 0; dd < DC; dd += 4) {
            const int col = dd + hi * 2;
            // A fragment: 16x4 fp32, row m, K = col, col+1 (8B-aligned ds_load_b64)
            v2f a = *(const v2f*)&tile[(rowBase + m) * LDS_PITCH + col];
            // B fragment: 4x16, every column = w chunk (broadcast)
            v2f bf;
            bf.x = wl[d0 + col];
            bf.y = wl[d0 + col + 1];
            acc = __builtin_amdgcn_wmma_f32_16x16x4_f32(
                      false, a, false, bf, (short)0, acc, false, false);
        }
    }

    // D column 0 lives in lane 0 (rows 0-7) and lane 16 (rows 8-15)
    if (m == 0) {
        const float bv = bias[0];
        const int rbase = k0 + rowBase + hi * 8;
        for (int r = 0; r < 8; ++r) {
            const int krow = rbase + r;
            float s = acc[r] + bv;
            if (mask[bi * KD + krow] == 0) s = NEG_INF_F;
            scores[(size_t)bi * KD + krow] = s;
        }
    }
}

// ---------------------------------------------------------------------------
// Kernel 2: per-row top-k via wave32 argmax butterfly. One wave per batch row.
// Also writes the 1/k indicator matrix (zero-initializing the poisoned output).
// ---------------------------------------------------------------------------
__global__ __launch_bounds__(128) void topk_kernel(
    const float* __restrict__ scores, const int* __restrict__ topk_p,
    float* __restrict__ attn, int* __restrict__ idx_out)
{
    const int lane = threadIdx.x & 31;
    const int wave = threadIdx.x >> 5;
    const int b    = blockIdx.x * 4 + wave;

    int kk = topk_p[0];
    if (kk > KD) kk = KD;
    if (kk > 32) kk = 32;
    const float invk = 1.0f / (float)kk;

    float s[16];
    unsigned used = 0;
    const float* srow = scores + (size_t)b * KD;
    for (int j = 0; j < 16; ++j) s[j] = srow[j * 32 + lane];

    for (int t = 0; t < kk; ++t) {
        float bv = -3.0e38f;
        int   bj = -1;
        for (int j = 0; j < 16; ++j) {
            if (!(used & (1u << j)) && s[j] > bv) { bv = s[j]; bj = j; }
        }
        int bidx = (bj >= 0) ? (bj * 32 + lane) : 0x7fffffff;
        // wave32 butterfly argmax; lower index wins ties (matches top_k)
        for (int off = 16; off >= 1; off >>= 1) {
            float ov = __shfl_xor(bv, off, 32);
            int   oi = __shfl_xor(bidx, off, 32);
            if (ov > bv || (ov == bv && oi < bidx)) { bv = ov; bidx = oi; }
        }
        if ((bidx & 31) == lane) used |= 1u << (bidx >> 5);
        if (lane == 0) idx_out[b * 32 + t] = bidx;
    }

    for (int j = 0; j < 16; ++j)
        attn[(size_t)b * KD + j * 32 + lane] = (used & (1u << j)) ? invk : 0.0f;
}

// ---------------------------------------------------------------------------
// Kernel 3: pooled[b,:] = mean of the k selected rows (gather hits L2).
// ---------------------------------------------------------------------------
__global__ __launch_bounds__(256) void pool_kernel(
    const float* __restrict__ emb, const int* __restrict__ idx,
    const int* __restrict__ topk_p, float* __restrict__ pooled)
{
    const int b = blockIdx.x;
    int kk = topk_p[0];
    if (kk > KD) kk = KD;
    if (kk > 32) kk = 32;
    const float invk = 1.0f / (float)kk;

    for (int d = threadIdx.x; d < DD; d += 256) {
        float acc = 0.0f;
        for (int j = 0; j < kk; ++j) {
            const int kr = idx[b * 32 + j];
            acc += emb[((size_t)b * KD + kr) * DD + d];
        }
        pooled[(size_t)b * DD + d] = acc * invk;
    }
}

// ---------------------------------------------------------------------------
extern "C" void kernel_launch(void* const* d_in, const int* in_sizes, int n_in,
                              void* d_out, int out_size, void* d_ws, size_t ws_size,
                              hipStream_t stream) {
    (void)in_sizes; (void)n_in; (void)out_size; (void)ws_size;

    const float* emb   = (const float*)d_in[0];   // (B, K, D) fp32
    const int*   mask  = (const int*)  d_in[1];   // (B, K) int32
    const float* w     = (const float*)d_in[2];   // (D,) fp32
    const float* bias  = (const float*)d_in[3];   // (1,) fp32
    const int*   topk  = (const int*)  d_in[4];   // scalar

    float* pooled = (float*)d_out;                // (B, D)
    float* attn   = (float*)d_out + (size_t)BB * DD;  // (B, K)

    float* scores = (float*)d_ws;                                  // B*K floats
    int*   idx    = (int*)((char*)d_ws + (size_t)BB * KD * sizeof(float)); // B*32 ints

    dim3 g1(KD / ROWS_PER_BLOCK, BB);             // (4, 128)
    scores_kernel<<<g1, 256, 0, stream>>>(emb, mask, w, bias, scores);
    topk_kernel<<<BB / 4, 128, 0, stream>>>(scores, topk, attn, idx);
    pool_kernel<<<BB, 256, 0, stream>>>(emb, idx, topk, pooled);
}